// KCompetitiveLayer_8839042695243
// MI455X (gfx1250) — compile-verified
//
#include <hip/hip_runtime.h>
#include <stdint.h>

typedef __attribute__((ext_vector_type(4))) float f32x4;

#define D_DIM 4096
#define BLK 256
#define ROWS_PER_BLK 4
#define ALPHA_C 6.26f

// Issue one row (16KB) as 4 async b128 copies per thread: global -> LDS.
// CDNA5 path: GLOBAL_LOAD_ASYNC_TO_LDS_B128 (GVS mode), tracked by ASYNCcnt.
// Input is streamed (read-once): mark non-temporal so it doesn't displace L2.
__device__ __forceinline__ void issue_async_row(const float* rowp, uint32_t lds_off,
                                                uint32_t tid) {
#pragma unroll
  for (int c = 0; c < 4; ++c) {
    uint32_t goff = (uint32_t)(c * BLK + tid) * 16u;  // byte offset in row
    uint32_t loff = lds_off + goff;                   // LDS byte address
    asm volatile("global_load_async_to_lds_b128 %0, %1, %2 th:TH_LOAD_NT"
                 :: "v"(loff), "v"(goff), "s"(rowp)
                 : "memory");
  }
}

__global__ void __launch_bounds__(BLK) kcomp_kernel(const float* __restrict__ x,
                                                    float* __restrict__ out, int B) {
  __shared__ f32x4 sh[2][D_DIM / 4];               // 2 x 16KB row buffers
  __shared__ unsigned long long s_red[2][8][2];    // parity / wave / {max,min}
  __shared__ float s_sum[8][2];

  const uint32_t tid  = threadIdx.x;
  const uint32_t lane = tid & 31u;
  const uint32_t wid  = tid >> 5;
  const int row0 = blockIdx.x * ROWS_PER_BLK;

  // addrspace(3) offsets are the low 32 bits of the flat LDS address
  const uint32_t lds_base = (uint32_t)(uintptr_t)(&sh[0][0]);

  if (row0 < B) issue_async_row(x + (size_t)row0 * D_DIM, lds_base, tid);

  for (int i = 0; i < ROWS_PER_BLK; ++i) {
    const int row = row0 + i;
    if (row >= B) break;
    const int buf = i & 1;

    // Wait for this wave's async copies, then sync all waves: buffer ready.
    asm volatile("s_wait_asynccnt 0x0" ::: "memory");
    __syncthreads();

    // LDS -> registers (ds_load_b128 x4)
    float v[16];
#pragma unroll
    for (int c = 0; c < 4; ++c) {
      f32x4 q = sh[buf][c * BLK + tid];
      v[c * 4 + 0] = q[0]; v[c * 4 + 1] = q[1];
      v[c * 4 + 2] = q[2]; v[c * 4 + 3] = q[3];
    }

    // Ensure our LDS reads retired, then prefetch next row into other buffer.
    asm volatile("s_wait_dscnt 0x0" ::: "memory");
    if (i + 1 < ROWS_PER_BLK && row + 1 < B)
      issue_async_row(x + (size_t)(row + 1) * D_DIM,
                      lds_base + (uint32_t)(buf ^ 1) * (D_DIM * 4u), tid);

    // Build unique order keys + branch sums.
    // okey = monotone-ordered float bits; composite key adds a 12-bit column
    // tiebreak (lower column wins on equal values, matching lax.top_k).
    unsigned long long ck[16];
    float sum_p = 0.f, sum_n = 0.f;
#pragma unroll
    for (int c = 0; c < 4; ++c) {
#pragma unroll
      for (int j = 0; j < 4; ++j) {
        const int e = c * 4 + j;
        const float val = v[e];
        const uint32_t col = (uint32_t)(c * BLK + tid) * 4u + (uint32_t)j;
        const uint32_t fb = __float_as_uint(val);
        const uint32_t ok = fb ^ ((uint32_t)((int32_t)fb >> 31) | 0x80000000u);
        ck[e] = ((unsigned long long)ok << 12) | (unsigned long long)(4095u - col);
        sum_p += fmaxf(val, 0.f);
        sum_n += fmaxf(-val, 0.f);
      }
    }

    // Block-wide row sums (wave32 shfl + one LDS round trip).
#pragma unroll
    for (int s = 16; s; s >>= 1) {
      sum_p += __shfl_xor(sum_p, s, 32);
      sum_n += __shfl_xor(sum_n, s, 32);
    }
    if (lane == 0) { s_sum[wid][0] = sum_p; s_sum[wid][1] = sum_n; }
    __syncthreads();
    sum_p = 0.f; sum_n = 0.f;
#pragma unroll
    for (int w = 0; w < 8; ++w) { sum_p += s_sum[w][0]; sum_n += s_sum[w][1]; }

    // 16 rounds of joint block max (< ub) and min (> lb): thresholds for
    // the 16 largest and 16 smallest unique keys (one barrier per round,
    // parity-double-buffered scratch).
    unsigned long long ub = ~0ull, lb = 0ull;
    for (int it = 0; it < 16; ++it) {
      unsigned long long cmax = 0ull, cmin = ~0ull;
#pragma unroll
      for (int e = 0; e < 16; ++e) {
        const unsigned long long k = ck[e];
        if (k < ub && k > cmax) cmax = k;
        const unsigned long long km = k ^ 0xFFFull;  // min-side tie flip
        if (km > lb && km < cmin) cmin = km;
      }
#pragma unroll
      for (int s = 16; s; s >>= 1) {
        unsigned long long o = __shfl_xor(cmax, s, 32); if (o > cmax) cmax = o;
        o = __shfl_xor(cmin, s, 32);                    if (o < cmin) cmin = o;
      }
      const int par = it & 1;
      if (lane == 0) { s_red[par][wid][0] = cmax; s_red[par][wid][1] = cmin; }
      __syncthreads();
      unsigned long long gmax = 0ull, gmin = ~0ull;
#pragma unroll
      for (int w = 0; w < 8; ++w) {
        const unsigned long long a = s_red[par][w][0]; if (a > gmax) gmax = a;
        const unsigned long long b = s_red[par][w][1]; if (b < gmin) gmin = b;
      }
      ub = gmax; lb = gmin;
    }
    const unsigned long long Thi = ub;   // 16th-largest composite key
    const unsigned long long Tlo = lb;   // 16th-smallest (min-space) key

    // Sums over the selected entries.
    float sel_p = 0.f, sel_n = 0.f;
#pragma unroll
    for (int e = 0; e < 16; ++e) {
      const unsigned long long k = ck[e];
      const float val = v[e];
      if (k >= Thi && val > 0.f) sel_p += val;
      if ((k ^ 0xFFFull) <= Tlo && val < 0.f) sel_n += -val;
    }
#pragma unroll
    for (int s = 16; s; s >>= 1) {
      sel_p += __shfl_xor(sel_p, s, 32);
      sel_n += __shfl_xor(sel_n, s, 32);
    }
    if (lane == 0) { s_sum[wid][0] = sel_p; s_sum[wid][1] = sel_n; }
    __syncthreads();
    sel_p = 0.f; sel_n = 0.f;
#pragma unroll
    for (int w = 0; w < 8; ++w) { sel_p += s_sum[w][0]; sel_n += s_sum[w][1]; }

    const float p_tmp = ALPHA_C * (sum_p - sel_p);
    const float n_tmp = ALPHA_C * (sum_n - sel_n);

    // Emit: x + p_tmp on top-16 positives, x - n_tmp on bottom-16 negatives,
    // 0 elsewhere. Streaming (nontemporal) b128 stores.
    float* orow = out + (size_t)row * D_DIM;
#pragma unroll
    for (int c = 0; c < 4; ++c) {
      f32x4 o;
#pragma unroll
      for (int j = 0; j < 4; ++j) {
        const int e = c * 4 + j;
        const float val = v[e];
        const unsigned long long k = ck[e];
        float r = 0.f;
        if (k >= Thi && val > 0.f)                    r = val + p_tmp;
        else if ((k ^ 0xFFFull) <= Tlo && val < 0.f)  r = val - n_tmp;
        o[j] = r;
      }
      __builtin_nontemporal_store(o, (f32x4*)(orow + (size_t)(c * BLK + tid) * 4));
    }
  }
}

extern "C" void kernel_launch(void* const* d_in, const int* in_sizes, int n_in,
                              void* d_out, int out_size, void* d_ws, size_t ws_size,
                              hipStream_t stream) {
  const float* x = (const float*)d_in[0];
  float* out = (float*)d_out;
  const int B = in_sizes[0] / D_DIM;                      // 8192
  const int grid = (B + ROWS_PER_BLK - 1) / ROWS_PER_BLK; // 2048 blocks
  hipLaunchKernelGGL(kcomp_kernel, dim3(grid), dim3(BLK), 0, stream, x, out, B);
}